// TnoBlock2d_57552561766743
// MI455X (gfx1250) — compile-verified
//
#include <hip/hip_runtime.h>
#include <hip/hip_bf16.h>

typedef __attribute__((ext_vector_type(2))) float v2f;
typedef __attribute__((ext_vector_type(8))) float v8f;

#define Bb 16
#define Cc 64
#define Hh 256
#define Wd 256
#define NN (Hh * Wd)          // 65536
#define BCN ((size_t)Bb * Cc * NN)

__device__ __forceinline__ v8f wmma4(v2f a, v2f b, v8f c) {
  // D = A(16x4) x B(4x16) + C(16x16), fp32 ; all modifiers 0 (F32: only CNeg legal)
  return __builtin_amdgcn_wmma_f32_16x16x4_f32(false, a, false, b, (short)0, c,
                                               false, false);
}

// exact DFT twiddles: angle = 2*pi*t/256 with t reduced mod 256
__device__ __forceinline__ float cos256(int t) {
  return __cosf(6.28318530717958647692f * (float)(t & 255) * (1.0f / 256.0f));
}
__device__ __forceinline__ float sin256(int t) {
  return __sinf(6.28318530717958647692f * (float)(t & 255) * (1.0f / 256.0f));
}

// Stage nFloats (multiple of 1024) from global to LDS via the gfx1250 async
// LDS-DMA path (GLOBAL_LOAD_ASYNC_TO_LDS_B128, tracked with ASYNCcnt).
// Assumes blockDim.x == 256 and 16B alignment of both pointers.
__device__ __forceinline__ void async_stage(float* lds, const float* gsrc,
                                            int nFloats) {
  int tid = threadIdx.x;
  for (int i = tid * 4; i < nFloats; i += 256 * 4) {
    unsigned ldsa = (unsigned)(unsigned long long)(lds + i);
    const float* g = gsrc + i;
    asm volatile("global_load_async_to_lds_b128 %0, %1, off"
                 :: "v"(ldsa), "v"(g) : "memory");
  }
  asm volatile("s_wait_asynccnt 0x0" ::: "memory");
  __syncthreads();
}

// ---------------------------------------------------------------------------
// Precompute DFT tables once into global scratch (stays L2-resident):
//  DFTWT[256][64]: forward rDFT along W (cos | -sin), scaled by 1/65536
//  IWT  [64][256]: inverse rDFT along W (s_c*cos | -s_c*sin), s_0=1 else 2
// ---------------------------------------------------------------------------
__global__ void k_tables(float* __restrict__ DFTWT, float* __restrict__ IWT) {
  int i = blockIdx.x * 256 + threadIdx.x;
  if (i < 256 * 64) {
    int w_ = i >> 6, c = i & 63;
    float v = (c < 32) ? cos256(w_ * c) : -sin256(w_ * (c - 32));
    DFTWT[i] = v * (1.0f / 65536.0f);
  } else {
    int j = i - 256 * 64;
    if (j < 64 * 256) {
      int c = j >> 8, w_ = j & 255;
      float v;
      if (c < 32) { float s = (c == 0) ? 1.0f : 2.0f; v = s * cos256(c * w_); }
      else { int ci = c - 32; float s = (ci == 0) ? 1.0f : 2.0f; v = -s * sin256(ci * w_); }
      IWT[j] = v;
    }
  }
}

// ---------------------------------------------------------------------------
// G1: partial rDFT along W.  F[b,i,h,c64] ; c<32: Re, c>=32: Im. norm 1/65536.
// A = x rows (M = B*C*H = 262144, K = 256), Bmat = DFT(256x64) in LDS.
// ---------------------------------------------------------------------------
__global__ void __launch_bounds__(256) k_dftW(const float* __restrict__ x,
                                              const float* __restrict__ DFTWT,
                                              float* __restrict__ F) {
  __shared__ __align__(16) float Bs[256 * 64];  // 64 KB
  async_stage(Bs, DFTWT, 256 * 64);
  int tid = threadIdx.x;
  int wave = tid >> 5, lane = tid & 31, l16 = lane & 15, half = lane >> 4;
  int job = blockIdx.x * 8 + wave;  // 65536 jobs
  int mBase = (job >> 2) * 16, nBase = (job & 3) * 16;
  const float* Arow = x + (size_t)(mBase + l16) * 256;
  v8f acc = {};
  for (int k = 0; k < 256; k += 4) {
    int ka = k + 2 * half;
    v2f a = *(const v2f*)(Arow + ka);
    v2f b;
    b.x = Bs[ka * 64 + nBase + l16];
    b.y = Bs[(ka + 1) * 64 + nBase + l16];
    acc = wmma4(a, b, acc);
  }
  float* Drow = F + (size_t)(mBase + 8 * half) * 64 + nBase + l16;
#pragma unroll
  for (int v = 0; v < 8; ++v) Drow[(size_t)v * 64] = acc[v];
}

// ---------------------------------------------------------------------------
// G2: DFT along H restricted to 64 retained rows (0..31, 224..255).
// Per (b,i): T1 = cosE*F, T2 = sinE*F (in LDS), then
//   X2r = T1[:, :32] + T2[:, 32:],  X2i = T1[:, 32:] - T2[:, :32]
// Twiddles built per-fragment with fast sincos (TRANS co-executes with WMMA).
// ---------------------------------------------------------------------------
__global__ void __launch_bounds__(256) k_dftH(const float* __restrict__ F,
                                              float* __restrict__ X2) {
  __shared__ float T[2][64 * 64];  // 32 KB
  int tid = threadIdx.x, wave = tid >> 5, lane = tid & 31;
  int l16 = lane & 15, half = lane >> 4;
  int bi = blockIdx.x;  // 0..1023
  const float* Fb = F + (size_t)bi * 256 * 64;
  for (int j = 0; j < 4; ++j) {
    int job = wave * 4 + j;           // 32 jobs
    int which = job >> 4;             // 0: cos, 1: sin
    int rBase = ((job >> 2) & 3) * 16, cBase = (job & 3) * 16;
    int r = rBase + l16;
    int ract = (r < 32) ? r : (r + 192);
    v8f acc = {};
    for (int k = 0; k < 256; k += 4) {
      int h = k + 2 * half;
      v2f a;
      if (which == 0) { a.x = cos256(ract * h); a.y = cos256(ract * (h + 1)); }
      else            { a.x = sin256(ract * h); a.y = sin256(ract * (h + 1)); }
      v2f b;
      b.x = Fb[h * 64 + cBase + l16];
      b.y = Fb[(h + 1) * 64 + cBase + l16];
      acc = wmma4(a, b, acc);
    }
    float* Td = T[which] + (rBase + 8 * half) * 64 + cBase + l16;
#pragma unroll
    for (int v = 0; v < 8; ++v) Td[v * 64] = acc[v];
  }
  __syncthreads();
  float* Xb = X2 + (size_t)bi * 64 * 64;
  for (int i = tid; i < 64 * 64; i += 256) {
    int r = i >> 6, c = i & 63;
    float o = (c < 32) ? (T[0][r * 64 + c] + T[1][r * 64 + c + 32])
                       : (T[0][r * 64 + c] - T[1][r * 64 + c - 32]);
    Xb[i] = o;
  }
}

// ---------------------------------------------------------------------------
// G3: per-mode complex channel mix. One block per (c,r); M=16 (batch), K=64.
// OutR = Xr*Wr - Xi*Wi ; OutI = Xr*Wi + Xi*Wr   (minus via VALU-negated A frag)
// ---------------------------------------------------------------------------
__global__ void __launch_bounds__(128) k_modemix(
    const float* __restrict__ X2, const float* __restrict__ w1r,
    const float* __restrict__ w1i, const float* __restrict__ w2r,
    const float* __restrict__ w2i, float* __restrict__ G3) {
  int tid = threadIdx.x, wave = tid >> 5, lane = tid & 31;
  int l16 = lane & 15, half = lane >> 4;
  int c = blockIdx.x;   // 0..31
  int r = blockIdx.y;   // 0..63
  const float *wr, *wi;
  int rm;
  if (r < 32) { wr = w1r; wi = w1i; rm = r; }
  else        { wr = w2r; wi = w2i; rm = r - 32; }
  int oBase = wave * 16;
  size_t wOff = (size_t)rm * 32 + c;
  v8f accR = {}, accI = {};
  for (int k = 0; k < 64; k += 4) {
    int i0 = k + 2 * half;
    size_t x0 = (((size_t)l16 * 64 + i0) * 64 + r) * 64 + c;
    v2f axr; axr.x = X2[x0];      axr.y = X2[x0 + 4096];
    v2f axi; axi.x = X2[x0 + 32]; axi.y = X2[x0 + 4096 + 32];
    v2f naxi; naxi.x = -axi.x; naxi.y = -axi.y;   // F32 WMMA has no neg_a
    size_t wb0 = ((size_t)i0 * 64 + oBase + l16) * 1024 + wOff;
    v2f bwr; bwr.x = wr[wb0]; bwr.y = wr[wb0 + 64 * 1024];
    v2f bwi; bwi.x = wi[wb0]; bwi.y = wi[wb0 + 64 * 1024];
    accR = wmma4(axr, bwr, accR);
    accR = wmma4(naxi, bwi, accR);   // - Xi*Wi
    accI = wmma4(axr, bwi, accI);
    accI = wmma4(axi, bwr, accI);
  }
#pragma unroll
  for (int v = 0; v < 8; ++v) {
    int b_ = v + 8 * half;
    size_t o0 = (((size_t)b_ * 64 + oBase + l16) * 64 + r) * 64 + c;
    G3[o0] = accR[v];
    G3[o0 + 32] = accI[v];
  }
}

// ---------------------------------------------------------------------------
// G4: inverse DFT along H (64 retained rows -> 256 h), e^{+i theta}.
// Y1r = cosE*Gr - sinE*Gi ; Y1i = cosE*Gi + sinE*Gr
// ---------------------------------------------------------------------------
__global__ void __launch_bounds__(256) k_invH(const float* __restrict__ G3,
                                              float* __restrict__ Y1) {
  __shared__ __align__(16) float Gs[64 * 64];  // 16 KB
  int tid = threadIdx.x, wave = tid >> 5, lane = tid & 31;
  int l16 = lane & 15, half = lane >> 4;
  int bo = blockIdx.x;  // 0..1023
  async_stage(Gs, G3 + (size_t)bo * 4096, 4096);
  float* Yb = Y1 + (size_t)bo * 256 * 64;
  for (int j = 0; j < 4; ++j) {
    int job = wave * 4 + j;           // 32 jobs: hTile(16) x ct(2)
    int hBase = (job >> 1) * 16, cB = (job & 1) * 16;
    int h = hBase + l16;
    v8f aCR = {}, aCI = {}, aSR = {}, aSI = {};
    for (int k = 0; k < 64; k += 4) {
      int r0 = k + 2 * half;
      int ra0 = (r0 < 32) ? r0 : r0 + 192;
      int ra1 = (r0 + 1 < 32) ? (r0 + 1) : (r0 + 193);
      v2f ac; ac.x = cos256(ra0 * h); ac.y = cos256(ra1 * h);
      v2f as; as.x = sin256(ra0 * h); as.y = sin256(ra1 * h);
      v2f bR; bR.x = Gs[r0 * 64 + cB + l16];      bR.y = Gs[(r0 + 1) * 64 + cB + l16];
      v2f bI; bI.x = Gs[r0 * 64 + cB + 32 + l16]; bI.y = Gs[(r0 + 1) * 64 + cB + 32 + l16];
      aCR = wmma4(ac, bR, aCR);
      aCI = wmma4(ac, bI, aCI);
      aSR = wmma4(as, bR, aSR);
      aSI = wmma4(as, bI, aSI);
    }
#pragma unroll
    for (int v = 0; v < 8; ++v) {
      int hh = hBase + v + 8 * half;
      Yb[(size_t)hh * 64 + cB + l16]      = aCR[v] - aSI[v];
      Yb[(size_t)hh * 64 + cB + 32 + l16] = aCI[v] + aSR[v];
    }
  }
}

// ---------------------------------------------------------------------------
// Skip: KQV[beta][b,o,n] = sum_i ws_beta[o,i]*x[b,i,n] + bias_beta[o]
// ---------------------------------------------------------------------------
__global__ void __launch_bounds__(256) k_skip(
    const float* __restrict__ x, const float* __restrict__ wsK,
    const float* __restrict__ bK, const float* __restrict__ wsQ,
    const float* __restrict__ bQ, const float* __restrict__ wsV,
    const float* __restrict__ bV, float* __restrict__ KQV) {
  int tid = threadIdx.x, wave = tid >> 5, lane = tid & 31;
  int l16 = lane & 15, half = lane >> 4;
  int b_ = blockIdx.y;
  int nBase0 = blockIdx.x * 64;
  const float* xb = x + (size_t)b_ * 64 * NN;
  for (int j = 0; j < 6; ++j) {
    int job = wave + j * 8;           // 48 jobs: beta(3) x oT(4) x nT(4)
    int beta = job >> 4, rem = job & 15;
    int oBase = (rem >> 2) * 16, nB = nBase0 + (rem & 3) * 16;
    const float* wsp = (beta == 0) ? wsK : (beta == 1) ? wsQ : wsV;
    const float* bp  = (beta == 0) ? bK  : (beta == 1) ? bQ  : bV;
    v8f acc = {};
    for (int k = 0; k < 64; k += 4) {
      int i0 = k + 2 * half;
      v2f a = *(const v2f*)(wsp + (oBase + l16) * 64 + i0);
      v2f b;
      b.x = xb[(size_t)i0 * NN + nB + l16];
      b.y = xb[(size_t)(i0 + 1) * NN + nB + l16];
      acc = wmma4(a, b, acc);
    }
    float* outp = KQV + (size_t)beta * BCN + (size_t)b_ * 64 * NN;
#pragma unroll
    for (int v = 0; v < 8; ++v) {
      int o = oBase + v + 8 * half;
      outp[(size_t)o * NN + nB + l16] = acc[v] + bp[o];
    }
  }
}

// ---------------------------------------------------------------------------
// G5: inverse rDFT along W (32 complex modes -> 256 real), accumulate into KQV.
// ---------------------------------------------------------------------------
__global__ void __launch_bounds__(256) k_accumW(const float* __restrict__ Y1,
                                                const float* __restrict__ IWT,
                                                float* __restrict__ out) {
  __shared__ __align__(16) float IW[64 * 256];  // 64 KB
  async_stage(IW, IWT, 64 * 256);
  int tid = threadIdx.x;
  int wave = tid >> 5, lane = tid & 31, l16 = lane & 15, half = lane >> 4;
  int mBase = blockIdx.x * 16;  // rows = (b*64+o)*256 + h
  const float* Ar = Y1 + (size_t)(mBase + l16) * 64;
  for (int j = 0; j < 2; ++j) {
    int wB = (wave * 2 + j) * 16;
    v8f acc = {};
    for (int k = 0; k < 64; k += 4) {
      int c0 = k + 2 * half;
      v2f a = *(const v2f*)(Ar + c0);
      v2f b;
      b.x = IW[c0 * 256 + wB + l16];
      b.y = IW[(c0 + 1) * 256 + wB + l16];
      acc = wmma4(a, b, acc);
    }
#pragma unroll
    for (int v = 0; v < 8; ++v) {
      size_t row = (size_t)(mBase + v + 8 * half);
      out[row * 256 + wB + l16] += acc[v];
    }
  }
}

__global__ void k_zero(float* __restrict__ p, int n) {
  int i = blockIdx.x * 256 + threadIdx.x;
  if (i < n) p[i] = 0.0f;
}

// ---------------------------------------------------------------------------
// Attention 1: DP[b,cq,ck] += (1/64) sum_n Q[b,cq,n]*K[b,ck,n] (split-K atomics)
// ---------------------------------------------------------------------------
__global__ void __launch_bounds__(256) k_qk(const float* __restrict__ Q,
                                            const float* __restrict__ Kk,
                                            float* __restrict__ DP) {
  int tid = threadIdx.x, wave = tid >> 5, lane = tid & 31;
  int l16 = lane & 15, half = lane >> 4;
  int b_ = blockIdx.y;
  int nBase = blockIdx.x * 1024;
  const float* Qb = Q + (size_t)b_ * 64 * NN;
  const float* Kb = Kk + (size_t)b_ * 64 * NN;
  for (int j = 0; j < 2; ++j) {
    int t = wave * 2 + j;  // 16 tiles: qT(4) x kT(4)
    int qT = t >> 2, kT = t & 3;
    const float* Arow = Qb + (size_t)(qT * 16 + l16) * NN + nBase;
    const float* Brow = Kb + (size_t)(kT * 16 + l16) * NN + nBase;
    v8f acc = {};
    for (int k = 0; k < 1024; k += 4) {
      int n0 = k + 2 * half;
      if ((k & 63) == 0) {
        __builtin_prefetch(Arow + n0 + 256, 0, 1);
        __builtin_prefetch(Brow + n0 + 256, 0, 1);
      }
      v2f a = *(const v2f*)(Arow + n0);
      v2f b = *(const v2f*)(Brow + n0);
      acc = wmma4(a, b, acc);
    }
#pragma unroll
    for (int v = 0; v < 8; ++v) {
      int cq = qT * 16 + v + 8 * half, ck = kT * 16 + l16;
      atomicAdd(&DP[((size_t)b_ * 64 + cq) * 64 + ck], acc[v] * (1.0f / 64.0f));
    }
  }
}

__global__ void k_softmax(float* __restrict__ DP) {
  int row = blockIdx.x * 256 + threadIdx.x;
  if (row >= Bb * 64) return;
  float* p = DP + (size_t)row * 64;
  float m = -1e30f;
  for (int i = 0; i < 64; ++i) m = fmaxf(m, p[i]);
  float s = 0.0f;
  for (int i = 0; i < 64; ++i) s += __expf(p[i] - m);
  float inv = 1.0f / s;
  for (int i = 0; i < 64; ++i) p[i] = __expf(p[i] - m) * inv;
}

// ---------------------------------------------------------------------------
// Attention 2: out[b,c,n] = sum_d DP[b,c,d]*V[b,d,n]
// ---------------------------------------------------------------------------
__global__ void __launch_bounds__(256) k_av(const float* __restrict__ DP,
                                            const float* __restrict__ V,
                                            float* __restrict__ out) {
  __shared__ __align__(16) float Ps[64 * 64];  // 16 KB
  int tid = threadIdx.x, wave = tid >> 5, lane = tid & 31;
  int l16 = lane & 15, half = lane >> 4;
  int b_ = blockIdx.y;
  int nBase0 = blockIdx.x * 256;
  async_stage(Ps, DP + (size_t)b_ * 4096, 4096);
  const float* Vb = V + (size_t)b_ * 64 * NN;
  float* Ob = out + (size_t)b_ * 64 * NN;
  for (int j = 0; j < 8; ++j) {
    int t = wave * 8 + j;  // 64 tiles: cT(4) x nT(16)
    int cT = t >> 4, nB = nBase0 + (t & 15) * 16;
    v8f acc = {};
    for (int k = 0; k < 64; k += 4) {
      int d0 = k + 2 * half;
      v2f a = *(const v2f*)(Ps + (cT * 16 + l16) * 64 + d0);
      v2f b;
      b.x = Vb[(size_t)d0 * NN + nB + l16];
      b.y = Vb[(size_t)(d0 + 1) * NN + nB + l16];
      acc = wmma4(a, b, acc);
    }
#pragma unroll
    for (int v = 0; v < 8; ++v) {
      int c = cT * 16 + v + 8 * half;
      Ob[(size_t)c * NN + nB + l16] = acc[v];
    }
  }
}

extern "C" void kernel_launch(void* const* d_in, const int* in_sizes, int n_in,
                              void* d_out, int out_size, void* d_ws,
                              size_t ws_size, hipStream_t stream) {
  const float* x = (const float*)d_in[0];
  // params dict order: K(w1r,w1i,w2r,w2i,ws,b), Q(...), V(...)
  const float* W[3][4];
  const float* WS[3];
  const float* BS[3];
  for (int br = 0; br < 3; ++br) {
    int base = 1 + br * 6;
    for (int t = 0; t < 4; ++t) W[br][t] = (const float*)d_in[base + t];
    WS[br] = (const float*)d_in[base + 4];
    BS[br] = (const float*)d_in[base + 5];
  }
  float* ws = (float*)d_ws;
  float* F     = ws;                     // 16,777,216 floats
  float* X2    = F + 16777216;           //  4,194,304
  float* G3    = X2 + 4194304;           //  4,194,304
  float* Y1    = G3 + 4194304;           // 16,777,216
  float* KQV   = Y1 + 16777216;          // 3 * 67,108,864
  float* DP    = KQV + 3 * BCN;          //     65,536
  float* DFTWT = DP + 65536;             //     16,384
  float* IWT   = DFTWT + 16384;          //     16,384
  float* out = (float*)d_out;

  // DFT tables (L2-resident thereafter)
  k_tables<<<128, 256, 0, stream>>>(DFTWT, IWT);
  // Shared spectral front-end
  k_dftW<<<8192, 256, 0, stream>>>(x, DFTWT, F);
  k_dftH<<<1024, 256, 0, stream>>>(F, X2);
  // Skip path (all three branches)
  k_skip<<<dim3(1024, 16), 256, 0, stream>>>(x, WS[0], BS[0], WS[1], BS[1],
                                             WS[2], BS[2], KQV);
  // Per-branch spectral back-end (K=0, Q=1, V=2), reusing G3/Y1 scratch
  for (int br = 0; br < 3; ++br) {
    k_modemix<<<dim3(32, 64), 128, 0, stream>>>(X2, W[br][0], W[br][1],
                                                W[br][2], W[br][3], G3);
    k_invH<<<1024, 256, 0, stream>>>(G3, Y1);
    k_accumW<<<16384, 256, 0, stream>>>(Y1, IWT, KQV + (size_t)br * BCN);
  }
  // Channel attention: dprod = softmax(q k^T / C); out = dprod v
  k_zero<<<(Bb * 64 * 64 + 255) / 256, 256, 0, stream>>>(DP, Bb * 64 * 64);
  k_qk<<<dim3(64, 16), 256, 0, stream>>>(KQV + BCN /*Q*/, KQV /*K*/, DP);
  k_softmax<<<4, 256, 0, stream>>>(DP);
  k_av<<<dim3(256, 16), 256, 0, stream>>>(DP, KQV + 2 * BCN /*V*/, out);
}